// GRU2_49658411876621
// MI455X (gfx1250) — compile-verified
//
#include <hip/hip_runtime.h>
#include <hip/hip_bf16.h>

typedef __attribute__((ext_vector_type(2))) float v2f;
typedef __attribute__((ext_vector_type(8))) float v8f;

#define B_ 64
#define T_ 1024
#define D_ 256
#define H_ 256
#define G_ 768 /* 3H */

// Native fp32 WMMA: D(16x16) = A(16x4) * B(4x16) + C, exact fp32 accumulate.
__device__ __forceinline__ v8f wmma_f32_k4(v2f a, v2f b, v8f c) {
    return __builtin_amdgcn_wmma_f32_16x16x4_f32(
        /*neg_a=*/false, a, /*neg_b=*/false, b,
        /*c_mod=*/(short)0, c, /*reuse_a=*/false, /*reuse_b=*/false);
}

// ---------------------------------------------------------------------------
// Phase 1: gates[dir][bt][g] = x[bt,:] . w_ih[dir][g,:] + b_ih[dir][g]
// Block = 512 threads (16 waves) -> M=64 x N=768 strip, K=256.
// A panel (64x256 fp32, 64 KB) staged in LDS once; each wave computes a
// 4(M) x 3(N) register tile block = 12 WMMA accumulators.
// ---------------------------------------------------------------------------
__global__ void gru_input_gemm(const float* __restrict__ x,
                               const float* __restrict__ w_ih_fw,
                               const float* __restrict__ b_ih_fw,
                               const float* __restrict__ w_ih_bw,
                               const float* __restrict__ b_ih_bw,
                               float* __restrict__ gates) {
    __shared__ float As[64 * D_];                       // 64 KB
    const int lane = threadIdx.x & 31;
    const int wave = threadIdx.x >> 5;                  // 0..15
    const int dir  = blockIdx.y;
    const long m0  = (long)blockIdx.x * 64;

    const float* w    = dir ? w_ih_bw : w_ih_fw;
    const float* bias = dir ? b_ih_bw : b_ih_fw;
    float*       out  = gates + (long)dir * (long)(B_ * T_) * G_;

    // Stage A panel: 64 rows x 256 cols, coalesced float4 copies.
    {
        const float4* xg  = (const float4*)(x + m0 * D_);
        float4*       as4 = (float4*)As;
        for (int i = threadIdx.x; i < 64 * (D_ / 4); i += blockDim.x)
            as4[i] = xg[i];
    }
    __syncthreads();

    const int rowl = lane & 15;        // m for A-frag / n for B-frag
    const int kk   = (lane >> 4) * 2;  // k sub-offset per half-wave
    const int n0   = wave * 48;        // this wave's first N column (3 tiles)

    v8f acc[4][3] = {};
    const float* Bc0 = w + (long)(n0 + rowl) * D_;      // j-th tile at +16*D_

#pragma unroll 4
    for (int k = 0; k < D_; k += 4) {
        v2f a[4];
#pragma unroll
        for (int mi = 0; mi < 4; ++mi) {
            const float* Ar = As + (mi * 16 + rowl) * D_ + k + kk;
            a[mi].x = Ar[0];
            a[mi].y = Ar[1];
        }
#pragma unroll
        for (int j = 0; j < 3; ++j) {
            v2f b;
            b.x = Bc0[j * 16 * D_ + k + kk];
            b.y = Bc0[j * 16 * D_ + k + kk + 1];
#pragma unroll
            for (int mi = 0; mi < 4; ++mi)
                acc[mi][j] = wmma_f32_k4(a[mi], b, acc[mi][j]);
        }
    }

    const int cm = (lane >> 4) * 8;    // C layout: m = r + 8*(lane/16)
    const int cn = lane & 15;
#pragma unroll
    for (int j = 0; j < 3; ++j) {
        const float bn = bias[n0 + j * 16 + cn];
#pragma unroll
        for (int mi = 0; mi < 4; ++mi) {
#pragma unroll
            for (int r = 0; r < 8; ++r)
                out[(m0 + mi * 16 + cm + r) * (long)G_ + n0 + j * 16 + cn] =
                    acc[mi][j][r] + bn;
        }
    }
}

// ---------------------------------------------------------------------------
// Phase 2: sequential scan. 8 blocks = 4 batch-chunks x 2 directions.
// Block = 512 threads = 16 waves; each wave owns 3 of the 48 N-tiles and
// shares one A-fragment of h (LDS) across them per K-step.
// ---------------------------------------------------------------------------
__global__ void gru_recurrence(const float* __restrict__ gates,
                               const float* __restrict__ w_hh_fw,
                               const float* __restrict__ b_hh_fw,
                               const float* __restrict__ w_hh_bw,
                               const float* __restrict__ b_hh_bw,
                               float* __restrict__ out) {
    __shared__ float h_lds[16 * H_];   // 16 KB
    __shared__ float gh_lds[16 * G_];  // 48 KB

    const int dir  = blockIdx.x & 1;
    const int bc   = blockIdx.x >> 1;
    const int b0   = bc * 16;
    const int lane = threadIdx.x & 31;
    const int wave = threadIdx.x >> 5;

    const float* w    = dir ? w_hh_bw : w_hh_fw;
    const float* bias = dir ? b_hh_bw : b_hh_fw;
    const float* gx   = gates + (long)dir * (long)(B_ * T_) * G_;
    float*       y    = out;                                          // (B,T,2H)
    float*       hlast = out + (long)B_ * T_ * 2 * H_ + (long)dir * B_ * H_;

    for (int i = threadIdx.x; i < 16 * H_; i += blockDim.x) h_lds[i] = 0.f;
    __syncthreads();

    const int rowl = lane & 15;
    const int kk   = (lane >> 4) * 2;
    const int n0   = wave * 48;                       // 3 tiles per wave
    const float* Bc0 = w + (long)(n0 + rowl) * H_;

    for (int s = 0; s < T_; ++s) {
        const int t = dir ? (T_ - 1 - s) : s;

        // gh = h @ w_hh^T : A-frag shared across the wave's 3 N-tiles.
        v8f acc[3] = {};
        const float* Arow = h_lds + rowl * H_;
#pragma unroll 8
        for (int k = 0; k < H_; k += 4) {
            v2f a;
            a.x = Arow[k + kk];
            a.y = Arow[k + kk + 1];
#pragma unroll
            for (int j = 0; j < 3; ++j) {
                v2f b;
                b.x = Bc0[j * 16 * H_ + k + kk];
                b.y = Bc0[j * 16 * H_ + k + kk + 1];
                acc[j] = wmma_f32_k4(a, b, acc[j]);
            }
        }
        {
            const int cm = (lane >> 4) * 8;
            const int cn = lane & 15;
#pragma unroll
            for (int j = 0; j < 3; ++j)
#pragma unroll
                for (int r = 0; r < 8; ++r)
                    gh_lds[(cm + r) * G_ + n0 + j * 16 + cn] = acc[j][r];
        }
        __syncthreads();

        // Elementwise gates + state update: 16x256 elements, 8 per thread.
        for (int i = threadIdx.x; i < 16 * H_; i += blockDim.x) {
            const int  rr = i >> 8;          // row within chunk (H_=256)
            const int  c  = i & (H_ - 1);    // hidden index
            const long bt = (long)(b0 + rr) * T_ + t;
            const float rx = gx[bt * G_ + c];
            const float ux = gx[bt * G_ + H_ + c];
            const float mx = gx[bt * G_ + 2 * H_ + c];
            const float rh = gh_lds[rr * G_ + c]          + bias[c];
            const float uh = gh_lds[rr * G_ + H_ + c]     + bias[H_ + c];
            const float mh = gh_lds[rr * G_ + 2 * H_ + c] + bias[2 * H_ + c];
            const float r = 1.f / (1.f + __expf(-(rx + rh)));
            const float z = 1.f / (1.f + __expf(-(ux + uh)));
            const float n = tanhf(mx + r * mh);
            const float hp = h_lds[rr * H_ + c];
            const float hn = z * hp + (1.f - z) * n;
            h_lds[rr * H_ + c] = hn;                      // (rr,c) owned by one thread
            y[bt * 2 * H_ + (long)dir * H_ + c] = hn;
        }
        __syncthreads();
    }

    for (int i = threadIdx.x; i < 16 * H_; i += blockDim.x) {
        const int rr = i >> 8;
        const int c  = i & (H_ - 1);
        hlast[(long)(b0 + rr) * H_ + c] = h_lds[rr * H_ + c];
    }
}

extern "C" void kernel_launch(void* const* d_in, const int* in_sizes, int n_in,
                              void* d_out, int out_size, void* d_ws, size_t ws_size,
                              hipStream_t stream) {
    const float* x       = (const float*)d_in[0];
    const float* w_ih_fw = (const float*)d_in[1];
    const float* w_hh_fw = (const float*)d_in[2];
    const float* b_ih_fw = (const float*)d_in[3];
    const float* b_hh_fw = (const float*)d_in[4];
    const float* w_ih_bw = (const float*)d_in[5];
    const float* w_hh_bw = (const float*)d_in[6];
    const float* b_ih_bw = (const float*)d_in[7];
    const float* b_hh_bw = (const float*)d_in[8];
    float* out   = (float*)d_out;
    float* gates = (float*)d_ws;   // 2 * B*T * 3H floats (fully overwritten)

    dim3 grid1((B_ * T_) / 64, 2);          // 1024 x 2 blocks, 512 threads each
    gru_input_gemm<<<grid1, 512, 0, stream>>>(x, w_ih_fw, b_ih_fw, w_ih_bw, b_ih_bw, gates);

    gru_recurrence<<<8, 512, 0, stream>>>(gates, w_hh_fw, b_hh_fw, w_hh_bw, b_hh_bw, out);
}